// soft_cldice_26018911879198
// MI455X (gfx1250) — compile-verified
//
#include <hip/hip_runtime.h>
#include <hip/hip_bf16.h>

typedef __attribute__((ext_vector_type(2))) float v2f;
typedef __attribute__((ext_vector_type(8))) float v8f;

static constexpr unsigned W  = 160, H = 160, D = 64, B = 2;
static constexpr unsigned SW = 1, SH = W, SD = W * H;        // element strides
static constexpr unsigned NN = B * D * H * W;                // 3,276,800 per tensor
static constexpr unsigned NT = 2u * NN;                      // pred + true slots
static constexpr int      TPB  = 256;
static constexpr unsigned EBLK = NT / TPB;                   // exact: 25600 blocks
static constexpr int      RBLK = 256;                        // reduction blocks
static constexpr unsigned NCHUNK = NN / 64u;                 // 51,200 64-elem chunks
static constexpr unsigned NWAVES = (unsigned)RBLK * (TPB / 32); // 2048 waves

__device__ __forceinline__ void decomp(unsigned j, unsigned& w, unsigned& h, unsigned& d) {
  w = j % W;
  unsigned r = j / W;
  h = r % H;
  d = (r / H) % D;   // batch dim sits above D; % D keeps neighbor checks batch-local
}

// soft_erode (clamped 7-point plus-shape min), with optional fused skel update
// carrying the PREVIOUS iteration's conditional accumulate:
//   accum condition (active_old && cur) == active_new, which is what `active`
//   holds by the time this kernel runs (step_flags folded into maxd_delta_k).
__global__ void erode_accum_k(const float* __restrict__ s0, const float* __restrict__ s1,
                              float* __restrict__ dst,
                              const float* __restrict__ toadd, float* __restrict__ skel,
                              const int* __restrict__ active, int doAccum) {
  unsigned i = blockIdx.x * (unsigned)TPB + threadIdx.x;
  unsigned t = (i >= NN) ? 1u : 0u;
  if (!active[t]) return;                 // gated off -> no erode, no accum (correct)
  if (doAccum) skel[i] += toadd[i];
  const float* s = t ? s1 : s0;
  unsigned j = i - t * NN;
  unsigned w, h, d;
  decomp(j, w, h, d);
  float m = s[j];
  if (w > 0)     m = fminf(m, s[j - SW]);
  if (w < W - 1) m = fminf(m, s[j + SW]);
  if (h > 0)     m = fminf(m, s[j - SH]);
  if (h < H - 1) m = fminf(m, s[j + SH]);
  if (d > 0)     m = fminf(m, s[j - SD]);
  if (d < D - 1) m = fminf(m, s[j + SD]);
  dst[i] = m;
}

// One separable pass of the 3x3x3 clamped max (used for the W and H axes).
template <unsigned DIV, unsigned SIZE, unsigned STRIDE>
__global__ void pool1d_max_k(const float* __restrict__ src, float* __restrict__ dst,
                             const int* __restrict__ active) {
  unsigned i = blockIdx.x * (unsigned)TPB + threadIdx.x;
  unsigned t = (i >= NN) ? 1u : 0u;
  if (!active[t]) return;
  unsigned j = i - t * NN;
  unsigned c = (j / DIV) % SIZE;
  float m = src[i];
  if (c > 0)        m = fmaxf(m, src[i - STRIDE]);
  if (c < SIZE - 1) m = fmaxf(m, src[i + STRIDE]);
  __builtin_prefetch(src + i + (unsigned)TPB * 64u, 0, 0);  // global_prefetch_b8
  dst[i] = m;
}

// Init phase: final dilate pass (D axis) fused with skel0 = relu(img - open(img)).
__global__ void maxd_skelinit_k(const float* __restrict__ src,
                                const float* __restrict__ s0, const float* __restrict__ s1,
                                float* __restrict__ skel) {
  unsigned i = blockIdx.x * (unsigned)TPB + threadIdx.x;
  unsigned t = (i >= NN) ? 1u : 0u;
  unsigned j = i - t * NN;
  unsigned d = (j / SD) % D;
  float m = src[i];
  if (d > 0)     m = fmaxf(m, src[i - SD]);
  if (d < D - 1) m = fmaxf(m, src[i + SD]);
  const float* s = t ? s1 : s0;
  skel[i] = fmaxf(s[j] - m, 0.f);
}

// Loop phase: final dilate pass (D axis) fused with
//   delta  = relu(im - open); to_add = relu(delta - skel*delta)
// plus block-any -> cur flag, plus last-block-done step_flags
// (active &= cur; cur = 0) so no separate tiny kernel is needed.
__global__ void maxd_delta_k(const float* __restrict__ src,   // after maxW,maxH
                             const float* __restrict__ im,
                             const float* __restrict__ skel,
                             float* __restrict__ toadd,
                             int* __restrict__ active, int* __restrict__ cur,
                             int* __restrict__ counter) {
  __shared__ int bf;
  unsigned i = blockIdx.x * (unsigned)TPB + threadIdx.x;
  unsigned t = (i >= NN) ? 1u : 0u;     // uniform per block (NN % TPB == 0)
  if (threadIdx.x == 0) bf = 0;
  __syncthreads();
  if (active[t]) {
    unsigned j = i - t * NN;
    unsigned d = (j / SD) % D;
    float m = src[i];
    if (d > 0)     m = fmaxf(m, src[i - SD]);
    if (d < D - 1) m = fmaxf(m, src[i + SD]);
    float delta = fmaxf(im[i] - m, 0.f);
    float ta = fmaxf(delta - skel[i] * delta, 0.f);
    toadd[i] = ta;
    if (ta > 0.f) bf = 1;               // race-benign: everyone writes 1
  }
  __syncthreads();
  if (threadIdx.x == 0) {
    if (bf) atomicOr(&cur[t], 1);
    __threadfence();
    int c = atomicAdd(counter, 1);      // every block increments, gated or not
    if (c == (int)gridDim.x - 1) {      // last finishing block: fold step_flags in
      int c0 = atomicOr(&cur[0], 0);    // read via atomic (device-coherent)
      int c1 = atomicOr(&cur[1], 0);
      active[0] = active[0] && c0;
      active[1] = active[1] && c1;
      cur[0] = 0; cur[1] = 0;
      *counter = 0;
      __threadfence();
    }
  }
}

__global__ void init_flags_k(int* __restrict__ active, int* __restrict__ cur,
                             int* __restrict__ counter) {
  int t = (int)threadIdx.x;
  if (t < 2) { active[t] = 1; cur[t] = 0; }
  if (t == 0) *counter = 0;
}

// Four global sums via V_WMMA_F32_16X16X4_F32 with A = ones(16x4):
// each WMMA folds a 64-float chunk into the f32 accumulator; row m=0
// (VGPR0, lanes 0..15) summed over lanes gives the chunk sum.
// Iteration 9's conditional skel += to_add is fused here: `active` gates are
// hoisted to float multipliers so all loads are unconditional packed b64
// (to_add values are finite by construction -> g*ta is NaN-safe).
__global__ void reduce_wmma_k(const float* __restrict__ skel,
                              const float* __restrict__ toadd,
                              const int* __restrict__ active,
                              const float* __restrict__ ypred,
                              const float* __restrict__ ytrue,
                              float* __restrict__ partials) {
  unsigned tid  = threadIdx.x;
  unsigned lane = tid & 31u;
  unsigned wid  = tid >> 5;
  unsigned gw   = blockIdx.x * (unsigned)(TPB / 32) + wid;
  const float gP = active[0] ? 1.f : 0.f;   // uniform, hoisted: no in-loop branches
  const float gT = active[1] ? 1.f : 0.f;

  const v2f* skP = (const v2f*)skel;
  const v2f* skT = (const v2f*)(skel + NN);
  const v2f* taP = (const v2f*)toadd;
  const v2f* taT = (const v2f*)(toadd + NN);
  const v2f* yt2 = (const v2f*)ytrue;
  const v2f* yp2 = (const v2f*)ypred;

  v8f c0 = {}, c1 = {}, c2 = {}, c3 = {};
  v2f a; a.x = 1.f; a.y = 1.f;

  // NCHUNK / NWAVES == 25 exactly: loop trip count uniform, EXEC all-ones.
  for (unsigned ch = gw; ch < NCHUNK; ch += NWAVES) {
    unsigned b2 = ch * 32u + lane;            // v2f index; 256B contiguous per wave
    v2f sp = skP[b2];
    v2f tp = taP[b2];
    v2f st = skT[b2];
    v2f tt = taT[b2];
    v2f yt = yt2[b2];
    v2f yp = yp2[b2];
    sp.x = fmaf(gP, tp.x, sp.x); sp.y = fmaf(gP, tp.y, sp.y);
    st.x = fmaf(gT, tt.x, st.x); st.y = fmaf(gT, tt.y, st.y);
    v2f b0 = sp * yt;                         // skel_pred * y_true
    v2f b2v = st * yp;                        // skel_true * y_pred
    c0 = __builtin_amdgcn_wmma_f32_16x16x4_f32(false, a, false, b0,  (short)0, c0, false, false);
    c1 = __builtin_amdgcn_wmma_f32_16x16x4_f32(false, a, false, sp,  (short)0, c1, false, false);
    c2 = __builtin_amdgcn_wmma_f32_16x16x4_f32(false, a, false, b2v, (short)0, c2, false, false);
    c3 = __builtin_amdgcn_wmma_f32_16x16x4_f32(false, a, false, st,  (short)0, c3, false, false);
  }

  __shared__ float red[TPB / 32][4][16];
  if (lane < 16u) {
    red[wid][0][lane] = c0[0];
    red[wid][1][lane] = c1[0];
    red[wid][2][lane] = c2[0];
    red[wid][3][lane] = c3[0];
  }
  __syncthreads();
  if (tid == 0) {   // fixed-order combine -> bitwise deterministic
    for (int k = 0; k < 4; ++k) {
      float s = 0.f;
      for (int wv = 0; wv < TPB / 32; ++wv)
        for (int l = 0; l < 16; ++l)
          s += red[wv][k][l];
      partials[blockIdx.x * 4 + k] = s;
    }
  }
}

__global__ void finalize_k(const float* __restrict__ partials, float* __restrict__ out) {
  float s0 = 0.f, s1 = 0.f, s2 = 0.f, s3 = 0.f;
  for (int bk = 0; bk < RBLK; ++bk) {
    s0 += partials[bk * 4 + 0];
    s1 += partials[bk * 4 + 1];
    s2 += partials[bk * 4 + 2];
    s3 += partials[bk * 4 + 3];
  }
  const float SM = 1e-6f;
  float tprec = (s0 + SM) / (s1 + SM);
  float tsens = (s2 + SM) / (s3 + SM);
  out[0] = 1.f - 2.f * (tprec * tsens) / (tprec + tsens);
}

extern "C" void kernel_launch(void* const* d_in, const int* in_sizes, int n_in,
                              void* d_out, int out_size, void* d_ws, size_t ws_size,
                              hipStream_t stream) {
  (void)in_sizes; (void)n_in; (void)out_size; (void)ws_size;
  const float* y_true = (const float*)d_in[0];
  const float* y_pred = (const float*)d_in[1];
  float* out = (float*)d_out;

  // Workspace layout (floats): 5 volumes of 2N, then partials, then int flags.
  float* base = (float*)d_ws;
  float* imA  = base;
  float* imB  = base + 1 * (size_t)NT;
  float* t1   = base + 2 * (size_t)NT;
  float* t2   = base + 3 * (size_t)NT;
  float* skel = base + 4 * (size_t)NT;
  float* partials = base + 5 * (size_t)NT;       // 4 * RBLK floats
  int* flags   = (int*)(partials + 4 * RBLK);
  int* active  = flags;
  int* cur     = flags + 2;
  int* counter = flags + 4;

  dim3 g(EBLK), blk(TPB);
  init_flags_k<<<1, 32, 0, stream>>>(active, cur, counter);

  // skel0 = relu(img - dilate(erode(img)))
  erode_accum_k<<<g, blk, 0, stream>>>(y_pred, y_true, t1, t2, skel, active, 0);
  pool1d_max_k<1,  W, SW><<<g, blk, 0, stream>>>(t1, t2, active);
  pool1d_max_k<SH, H, SH><<<g, blk, 0, stream>>>(t2, t1, active);
  maxd_skelinit_k<<<g, blk, 0, stream>>>(t1, y_pred, y_true, skel);

  const float* cur0 = y_pred;
  const float* cur1 = y_true;
  float* ping = imA;
  float* pong = imB;
  for (int it = 0; it < 10; ++it) {
    // im = erode(im); also applies previous iteration's gated skel += to_add
    erode_accum_k<<<g, blk, 0, stream>>>(cur0, cur1, ping, t2, skel, active, it > 0 ? 1 : 0);
    // e2 = erode(im)
    erode_accum_k<<<g, blk, 0, stream>>>(ping, ping + NN, t1, t2, skel, active, 0);
    // open = dilate(e2): maxW, maxH, then maxD fused with delta/to_add/flags
    pool1d_max_k<1,  W, SW><<<g, blk, 0, stream>>>(t1, t2, active);
    pool1d_max_k<SH, H, SH><<<g, blk, 0, stream>>>(t2, t1, active);
    maxd_delta_k<<<g, blk, 0, stream>>>(t1, ping, skel, t2, active, cur, counter);
    cur0 = ping; cur1 = ping + NN;
    float* tmp = ping; ping = pong; pong = tmp;
  }

  // Iteration 9's gated accumulate is fused into the reduction.
  reduce_wmma_k<<<RBLK, TPB, 0, stream>>>(skel, t2, active, y_pred, y_true, partials);
  finalize_k<<<1, 1, 0, stream>>>(partials, out);
}